// Net_68968584839192
// MI455X (gfx1250) — compile-verified
//
#include <hip/hip_runtime.h>
#include <hip/hip_bf16.h>

// ---- problem sizes (fixed by the reference) ----
#define NN 100000      // nodes
#define NE 3200000     // edges (without self loops)
#define NF 512         // input features
#define NH 16          // hidden
#define NC 64          // classes

typedef __attribute__((ext_vector_type(2))) float v2f;
typedef __attribute__((ext_vector_type(8))) float v8f;

// ---------------------------------------------------------------------------
// init: deg = 1 (self loop), agg1 = 0, agg2 = 0.   64*NN threads exactly.
// ---------------------------------------------------------------------------
__global__ void k_init(float* __restrict__ deg, float* __restrict__ agg1,
                       float* __restrict__ agg2) {
    size_t i = (size_t)blockIdx.x * 256 + threadIdx.x;   // < 64*NN
    agg2[i] = 0.0f;
    if (i < (size_t)16 * NN) agg1[i] = 0.0f;
    if (i < (size_t)NN)      deg[i]  = 1.0f;
}

// ---------------------------------------------------------------------------
// degree accumulation over real edges (self loop already counted via init=1)
// ---------------------------------------------------------------------------
__global__ void k_deg(const long long* __restrict__ dst, float* __restrict__ deg) {
    size_t e = (size_t)blockIdx.x * 256 + threadIdx.x;   // < NE exactly
    long long d = dst[e];
    __hip_atomic_fetch_add(&deg[d], 1.0f, __ATOMIC_RELAXED, __HIP_MEMORY_SCOPE_AGENT);
}

// dinv = rsqrt(max(deg,1)) in place
__global__ void k_dinv(float* __restrict__ deg) {
    size_t i = (size_t)blockIdx.x * 256 + threadIdx.x;
    if (i < (size_t)NN) deg[i] = __frsqrt_rn(fmaxf(deg[i], 1.0f));
}

// ---------------------------------------------------------------------------
// GEMM1: h1_pre = x @ W1   [NN x NH], K = NF.  WMMA f32 16x16x4.
// One wave per 16-row tile; 8 waves / block; W1 staged in LDS (32 KB).
// A layout (16x4 f32): lane = 16*(k>=2) + m, VGPR0/1 = 2 consecutive K.
// C/D layout: vgpr r, lanes 0-15 -> M=r, lanes 16-31 -> M=r+8.
// ---------------------------------------------------------------------------
__global__ void k_gemm1(const float* __restrict__ x, const float* __restrict__ W1,
                        float* __restrict__ h1p) {
    __shared__ float sW[NF * NH];                 // 8192 floats = 32 KB
    int tid = threadIdx.x;
    for (int i = tid; i < NF * NH; i += 256) sW[i] = W1[i];
    __syncthreads();

    int tile = blockIdx.x * 8 + (tid >> 5);       // 16-row tile index
    if (tile >= NN / 16) return;                  // wave-uniform guard
    int lane = tid & 31;
    int n    = lane & 15;                         // = m for A, col for B/C
    int kh   = lane >> 4;                         // K half (0: K0/1, 1: K2/3)

    const float* arow = x + (size_t)(tile * 16 + n) * NF + 2 * kh;
    v8f acc = {};
    for (int k0 = 0; k0 < NF; k0 += 4) {
        v2f a = *(const v2f*)(arow + k0);
        v2f b;
        b.x = sW[(k0 + 2 * kh)     * NH + n];
        b.y = sW[(k0 + 2 * kh + 1) * NH + n];
        acc = __builtin_amdgcn_wmma_f32_16x16x4_f32(
            false, a, false, b, (short)0, acc, false, false);
    }
    float* orow = h1p + (size_t)(tile * 16 + 8 * kh) * NH + n;
#pragma unroll
    for (int r = 0; r < 8; ++r) orow[r * NH] = acc[r];
}

// ---------------------------------------------------------------------------
// scatter 1: agg1[dst] += h1_pre[src] * dinv[src]*dinv[dst]  (16 feat/edge)
// ---------------------------------------------------------------------------
__global__ void k_scatter1(const long long* __restrict__ src,
                           const long long* __restrict__ dst,
                           const float* __restrict__ dinv,
                           const float* __restrict__ h1p,
                           float* __restrict__ agg1) {
    size_t idx = (size_t)blockIdx.x * 256 + threadIdx.x;  // < NE*16 exactly
    size_t e = idx >> 4;
    int    f = (int)(idx & 15);
    long long s = src[e], d = dst[e];
    float nrm = dinv[s] * dinv[d];
    float msg = h1p[(size_t)s * NH + f] * nrm;
    __hip_atomic_fetch_add(&agg1[(size_t)d * NH + f], msg,
                           __ATOMIC_RELAXED, __HIP_MEMORY_SCOPE_AGENT);
}

// finalize layer 1: h1 = relu(agg1 + selfloop + b1)   (in place into agg1)
__global__ void k_final1(float* __restrict__ agg1, const float* __restrict__ h1p,
                         const float* __restrict__ dinv, const float* __restrict__ b1) {
    size_t idx = (size_t)blockIdx.x * 256 + threadIdx.x;  // < NN*16 exactly
    size_t nidx = idx >> 4;
    int    f    = (int)(idx & 15);
    float di = dinv[nidx];
    float v = agg1[idx] + h1p[idx] * di * di + b1[f];
    agg1[idx] = fmaxf(v, 0.0f);
}

// ---------------------------------------------------------------------------
// GEMM2: h2_pre = h1 @ W2   [NN x NC], K = NH.  4 column tiles of 16.
// ---------------------------------------------------------------------------
__global__ void k_gemm2(const float* __restrict__ h1, const float* __restrict__ W2,
                        float* __restrict__ h2p) {
    __shared__ float sW[NH * NC];                 // 1024 floats = 4 KB
    int tid = threadIdx.x;
    for (int i = tid; i < NH * NC; i += 256) sW[i] = W2[i];
    __syncthreads();

    int wt = blockIdx.x * 8 + (tid >> 5);         // 0 .. 25000-1 exactly
    int rt = wt >> 2;                             // row tile
    int ct = wt & 3;                              // 16-col tile
    int lane = tid & 31;
    int n    = lane & 15;
    int kh   = lane >> 4;

    const float* arow = h1 + (size_t)(rt * 16 + n) * NH + 2 * kh;
    v8f acc = {};
#pragma unroll
    for (int k0 = 0; k0 < NH; k0 += 4) {
        v2f a = *(const v2f*)(arow + k0);
        v2f b;
        b.x = sW[(k0 + 2 * kh)     * NC + ct * 16 + n];
        b.y = sW[(k0 + 2 * kh + 1) * NC + ct * 16 + n];
        acc = __builtin_amdgcn_wmma_f32_16x16x4_f32(
            false, a, false, b, (short)0, acc, false, false);
    }
    float* orow = h2p + (size_t)(rt * 16 + 8 * kh) * NC + ct * 16 + n;
#pragma unroll
    for (int r = 0; r < 8; ++r) orow[r * NC] = acc[r];
}

// ---------------------------------------------------------------------------
// scatter 2: agg2[dst] += h2_pre[src] * norm   (64 feat/edge)
// ---------------------------------------------------------------------------
__global__ void k_scatter2(const long long* __restrict__ src,
                           const long long* __restrict__ dst,
                           const float* __restrict__ dinv,
                           const float* __restrict__ h2p,
                           float* __restrict__ agg2) {
    size_t idx = (size_t)blockIdx.x * 256 + threadIdx.x;  // < NE*64 exactly
    size_t e = idx >> 6;
    int    f = (int)(idx & 63);
    long long s = src[e], d = dst[e];
    float nrm = dinv[s] * dinv[d];
    float msg = h2p[(size_t)s * NC + f] * nrm;
    __hip_atomic_fetch_add(&agg2[(size_t)d * NC + f], msg,
                           __ATOMIC_RELAXED, __HIP_MEMORY_SCOPE_AGENT);
}

// ---------------------------------------------------------------------------
// finalize layer 2 + log_softmax. One wave per node, 2 classes per lane.
// ---------------------------------------------------------------------------
__global__ void k_final2(const float* __restrict__ agg2, const float* __restrict__ h2p,
                         const float* __restrict__ dinv, const float* __restrict__ b2,
                         float* __restrict__ out) {
    int tid  = threadIdx.x;
    int lane = tid & 31;
    size_t node = (size_t)blockIdx.x * 8 + (tid >> 5);    // < NN exactly
    float di = dinv[node];
    float sn = di * di;
    const float* a  = agg2 + node * NC;
    const float* hp = h2p  + node * NC;
    float v0 = a[lane]      + hp[lane]      * sn + b2[lane];
    float v1 = a[lane + 32] + hp[lane + 32] * sn + b2[lane + 32];

    float m = fmaxf(v0, v1);
#pragma unroll
    for (int off = 16; off >= 1; off >>= 1) m = fmaxf(m, __shfl_xor(m, off, 32));
    float ssum = __expf(v0 - m) + __expf(v1 - m);
#pragma unroll
    for (int off = 16; off >= 1; off >>= 1) ssum += __shfl_xor(ssum, off, 32);
    float ls = __logf(ssum);

    out[node * NC + lane]      = v0 - m - ls;
    out[node * NC + lane + 32] = v1 - m - ls;
}

// ---------------------------------------------------------------------------
extern "C" void kernel_launch(void* const* d_in, const int* in_sizes, int n_in,
                              void* d_out, int out_size, void* d_ws, size_t ws_size,
                              hipStream_t stream) {
    const float*     x  = (const float*)d_in[0];
    const long long* ei = (const long long*)d_in[1];   // int64 per reference
    const float*     W1 = (const float*)d_in[2];
    const float*     b1 = (const float*)d_in[3];
    const float*     W2 = (const float*)d_in[4];
    const float*     b2 = (const float*)d_in[5];
    float* out = (float*)d_out;

    const long long* src = ei;           // edge_index[0]
    const long long* dst = ei + NE;      // edge_index[1]

    // workspace carve-up (floats): deg/dinv | h1_pre | agg1/h1 | h2_pre | agg2
    float* ws   = (float*)d_ws;
    float* deg  = ws;                                  // NN
    float* h1p  = ws + (size_t)NN;                     // 16*NN
    float* agg1 = ws + (size_t)NN * 17;                // 16*NN
    float* h2p  = ws + (size_t)NN * 33;                // 64*NN
    float* agg2 = ws + (size_t)NN * 97;                // 64*NN   (total 161*NN)

    k_init    <<<(64 * NN) / 256, 256, 0, stream>>>(deg, agg1, agg2);
    k_deg     <<<NE / 256, 256, 0, stream>>>(dst, deg);
    k_dinv    <<<(NN + 255) / 256, 256, 0, stream>>>(deg);

    k_gemm1   <<<(NN / 16 + 7) / 8, 256, 0, stream>>>(x, W1, h1p);
    k_scatter1<<<((size_t)NE * 16) / 256, 256, 0, stream>>>(src, dst, deg, h1p, agg1);
    k_final1  <<<(NN * 16) / 256, 256, 0, stream>>>(agg1, h1p, deg, b1);

    k_gemm2   <<<(NN / 16) * 4 / 8, 256, 0, stream>>>(agg1, W2, h2p);
    k_scatter2<<<((size_t)NE * 64) / 256, 256, 0, stream>>>(src, dst, deg, h2p, agg2);
    k_final2  <<<NN / 8, 256, 0, stream>>>(agg2, h2p, deg, b2, out);
}